// VanillaGRUModel_44461501448762
// MI455X (gfx1250) — compile-verified
//
#include <hip/hip_runtime.h>
#include <math.h>

typedef __attribute__((ext_vector_type(2))) float v2f;
typedef __attribute__((ext_vector_type(8))) float v8f;

#define B_ 1024
#define T_ 512
#define F_ 16
#define H_ 64
#define G_ 192   // 3*H

// LDS row strides padded for conflict-free fragment access (64 banks x 4B):
// stride 68 -> bank = (4*row + col) mod 64 (distinct across the 32-lane frag
// pattern), and 68*4 = 272 bytes keeps ds_load b64 8-byte aligned.
#define HS_ 68   // hbuf / zbuf / rbuf row stride (floats)
#define XS_ 20   // x-tile row stride (floats); 20*4 = 80 B, 8-aligned, perm banks

__device__ __forceinline__ float sigmoidf_(float v) {
    return 1.0f / (1.0f + __expf(-v));
}

// One workgroup = 12 wave32s = 384 threads, owns a 16-row batch tile (64 WGs).
// Wave w computes gate columns [16w,16w+16) of the 192-wide projections.
// U/W B-fragments + biases live in VGPRs for the whole kernel. Per step only a
// double-buffered 16x16 x tile and the 16x64 h state move through LDS, with
// 2 barriers per step and the next x tile prefetched behind the WMMA chain.
__global__ __launch_bounds__(384) void gru_fused_kernel(
    const float* __restrict__ x,      // [B,T,F]
    const float* __restrict__ W,      // [F,3H]
    const float* __restrict__ U,      // [H,3H]
    const float* __restrict__ b,      // [2,3H]
    const float* __restrict__ w1,     // [H,64]
    const float* __restrict__ b1,     // [64]
    const float* __restrict__ gamma,  // [64]
    const float* __restrict__ beta,   // [64]
    const float* __restrict__ mmean,  // [64]
    const float* __restrict__ mvar,   // [64]
    const float* __restrict__ w2,     // [64,1]
    const float* __restrict__ b2,     // [1]
    float* __restrict__ out)          // [B,1]
{
    __shared__ float xt[2][16 * XS_];  // double-buffered x_t tile
    __shared__ float hbuf[16 * HS_];   // hidden state
    __shared__ float zbuf[16 * HS_];   // update gate
    __shared__ float rbuf[16 * HS_];   // reset gate (reused as head staging)

    const int tid  = threadIdx.x;
    const int wave = tid >> 5;         // 0..11
    const int lane = tid & 31;
    const int nlo  = lane & 15;        // col-in-tile / A-row
    const int khi  = (lane >> 4) * 2;  // A/B frag k sub-offset: 0 or 2
    const int rhi  = (lane >> 4) * 8;  // C/D frag row offset: 0 or 8
    const int m0   = blockIdx.x * 16;  // batch tile base
    const int col  = wave * 16 + nlo;  // global gate column in [0,192)

    // --- persistent B-matrix fragments (V_WMMA_F32_16X16X4_F32 layout):
    // VGPR0 -> K=kb+0, VGPR1 -> K=kb+1; lanes 0-15 kb=4kk, lanes 16-31 kb=4kk+2.
    v2f Ufr[16];
#pragma unroll
    for (int kk = 0; kk < 16; ++kk) {
        int kb = kk * 4 + khi;
        v2f f = { U[(kb + 0) * G_ + col], U[(kb + 1) * G_ + col] };
        Ufr[kk] = f;
    }
    v2f Wfr[4];
#pragma unroll
    for (int kk = 0; kk < 4; ++kk) {
        int kb = kk * 4 + khi;
        v2f f = { W[(kb + 0) * G_ + col], W[(kb + 1) * G_ + col] };
        Wfr[kk] = f;
    }
    const float bi = b[col];       // input bias
    const float br = b[G_ + col];  // recurrent bias

    // init h = 0 and stage x_{t=0} into buffer 0
    for (int i = tid; i < 16 * HS_; i += 384) hbuf[i] = 0.0f;
    if (tid < 256) {
        int row = tid >> 4, c = tid & 15;
        xt[0][row * XS_ + c] = x[(size_t)(m0 + row) * (T_ * F_) + c];
    }
    __syncthreads();

    for (int t = 0; t < T_; ++t) {
        const int cur = t & 1;

        // A) issue global load of x_{t+1} early; latency hides behind WMMAs
        float xreg = 0.0f;
        if (tid < 256 && t + 1 < T_) {
            int row = tid >> 4, c = tid & 15;
            xreg = x[(size_t)(m0 + row) * (T_ * F_) + (size_t)(t + 1) * F_ + c];
        }

        // B) projections: xp = x_t@W + b_i (K=16), rc = h@U + b_r (K=64)
        v8f xp, rc;
#pragma unroll
        for (int i = 0; i < 8; ++i) { xp[i] = bi; rc[i] = br; }
#pragma unroll
        for (int kk = 0; kk < 4; ++kk) {
            int kb = kk * 4 + khi;
            v2f a = { xt[cur][nlo * XS_ + kb], xt[cur][nlo * XS_ + kb + 1] };
            xp = __builtin_amdgcn_wmma_f32_16x16x4_f32(
                false, a, false, Wfr[kk], (short)0, xp, false, false);
        }
#pragma unroll
        for (int kk = 0; kk < 16; ++kk) {
            int kb = kk * 4 + khi;
            v2f a = { hbuf[nlo * HS_ + kb], hbuf[nlo * HS_ + kb + 1] };
            rc = __builtin_amdgcn_wmma_f32_16x16x4_f32(
                false, a, false, Ufr[kk], (short)0, rc, false, false);
        }

        // C) publish gates + next x tile
        if (wave < 4) {            // z gate, columns [0,64)
#pragma unroll
            for (int i = 0; i < 8; ++i)
                zbuf[(i + rhi) * HS_ + col] = sigmoidf_(xp[i] + rc[i]);
        } else if (wave < 8) {     // r gate, columns [64,128)
#pragma unroll
            for (int i = 0; i < 8; ++i)
                rbuf[(i + rhi) * HS_ + (wave - 4) * 16 + nlo] = sigmoidf_(xp[i] + rc[i]);
        }
        if (tid < 256 && t + 1 < T_) {  // loadcnt wait lands here, after compute
            int row = tid >> 4, c = tid & 15;
            xt[cur ^ 1][row * XS_ + c] = xreg;
        }
        __syncthreads();  // barrier 1: z/r/x_next visible; hbuf reads fenced

        // D) in-place state update by waves 8-11 (hold xh=xp, rh=rc frags)
        if (wave >= 8) {
#pragma unroll
            for (int i = 0; i < 8; ++i) {
                int row = i + rhi;
                int c   = (wave - 8) * 16 + nlo;
                float rv = rbuf[row * HS_ + c];
                float zv = zbuf[row * HS_ + c];
                float hh = tanhf(xp[i] + rv * rc[i]);
                hbuf[row * HS_ + c] = zv * hbuf[row * HS_ + c] + (1.0f - zv) * hh;
            }
        }
        __syncthreads();  // barrier 2: new h visible for next step
    }

    // ---- head: y = relu(h@w1 + b1); BN(inference); out = y@w2 + b2
    if (wave < 4) {  // wave w owns y columns [16w,16w+16), col in [0,64)
        v8f acc;
#pragma unroll
        for (int i = 0; i < 8; ++i) acc[i] = b1[col];
#pragma unroll
        for (int kk = 0; kk < 16; ++kk) {
            int kb = kk * 4 + khi;
            v2f a  = { hbuf[nlo * HS_ + kb], hbuf[nlo * HS_ + kb + 1] };
            v2f bb = { w1[(kb + 0) * H_ + col], w1[(kb + 1) * H_ + col] };
            acc = __builtin_amdgcn_wmma_f32_16x16x4_f32(
                false, a, false, bb, (short)0, acc, false, false);
        }
        float sc = rsqrtf(mvar[col] + 1e-3f) * gamma[col];
#pragma unroll
        for (int i = 0; i < 8; ++i) {
            int row = i + rhi;
            float v = fmaxf(acc[i], 0.0f);
            v = (v - mmean[col]) * sc + beta[col];
            rbuf[row * HS_ + col] = v * w2[col];  // fold final matvec weight
        }
    }
    __syncthreads();
    if (tid < 16) {
        float s = b2[0];
        for (int c = 0; c < H_; ++c) s += rbuf[tid * HS_ + c];
        out[m0 + tid] = s;
    }
}

extern "C" void kernel_launch(void* const* d_in, const int* in_sizes, int n_in,
                              void* d_out, int out_size, void* d_ws, size_t ws_size,
                              hipStream_t stream) {
    (void)in_sizes; (void)n_in; (void)out_size; (void)d_ws; (void)ws_size;
    const float* x     = (const float*)d_in[0];
    const float* W     = (const float*)d_in[1];
    const float* U     = (const float*)d_in[2];
    const float* b     = (const float*)d_in[3];
    const float* w1    = (const float*)d_in[4];
    const float* b1    = (const float*)d_in[5];
    const float* gamma = (const float*)d_in[6];
    const float* beta  = (const float*)d_in[7];
    const float* mmean = (const float*)d_in[8];
    const float* mvar  = (const float*)d_in[9];
    const float* w2    = (const float*)d_in[10];
    const float* b2    = (const float*)d_in[11];
    float* out = (float*)d_out;

    gru_fused_kernel<<<B_ / 16, 384, 0, stream>>>(
        x, W, U, b, w1, b1, gamma, beta, mmean, mvar, w2, b2, out);
}